// GuidedMoEBasic_44212393345443
// MI455X (gfx1250) — compile-verified
//
#include <hip/hip_runtime.h>
#include <math.h>

typedef __attribute__((ext_vector_type(2))) float v2f;
typedef __attribute__((ext_vector_type(8))) float v8f;

#define BB   16
#define DD   64
#define HH   768
#define NEV  7
#define FEAT 776
#define NU   (BB*DD)          // 1024 utterances
#define PPB  (DD*(DD+1)/2)    // 2080 pairs per batch
#define NPAIR (BB*PPB)        // 33280
#define NC   16               // padded combined-projection columns (12 used)

// workspace layout (in floats)
#define WS_CONC 0                         // [1024][776]
#define WS_WCT  (NU*FEAT)                 // [16][776]  transposed combined weights
#define WS_P    (WS_WCT + NC*FEAT)        // [1024][16] projection result
#define WS_OB   (WS_P + NU*NC)            // [4] combined o-bias (e,c)

// ---------------------------------------------------------------------------
// Kernel 0: fold the affine expert MLP into per-utterance projection columns.
//   cols 0..1 : gate_w first-half (t)      cols 2..3 : gate_w second-half (end)
//   cols 4..7 : Mt[e][:,c] = W1[e][:776]@W2[e]   (e,c)=(0,0),(0,1),(1,0),(1,1)
//   cols 8..11: Me[e][:,c] = W1[e][776:]@W2[e]
//   cols 12..15: zero padding
// Also computes obias[e*2+c] = b1[e]@W2[e][:,c] + b2[e][c].
// ---------------------------------------------------------------------------
__global__ void k_build_weights(const float* __restrict__ gate_w,
                                const float* __restrict__ exp_w1,
                                const float* __restrict__ exp_b1,
                                const float* __restrict__ exp_w2,
                                const float* __restrict__ exp_b2,
                                float* __restrict__ ws) {
  int gid = blockIdx.x * blockDim.x + threadIdx.x;
  float* WcT = ws + WS_WCT;
  if (gid < NC * FEAT) {
    int c = gid / FEAT;
    int k = gid % FEAT;
    float v = 0.0f;
    if (c < 2) {
      v = gate_w[k * 2 + c];
    } else if (c < 4) {
      v = gate_w[(FEAT + k) * 2 + (c - 2)];
    } else if (c < 12) {
      int cc   = c - 4;
      int half = cc >> 2;          // 0 = t-half rows, 1 = end-half rows
      int e    = (cc >> 1) & 1;
      int col  = cc & 1;
      int krow = half ? (FEAT + k) : k;
      const float* w1 = exp_w1 + ((size_t)e * 2 * FEAT + krow) * 256;
      const float* w2 = exp_w2 + e * 512 + col;
      float s = 0.0f;
      for (int h = 0; h < 256; ++h) s += w1[h] * w2[h * 2];
      v = s;
    }
    WcT[c * FEAT + k] = v;
  }
  if (gid < 4) {
    int e = gid >> 1, col = gid & 1;
    float s = exp_b2[e * 2 + col];
    for (int h = 0; h < 256; ++h)
      s += exp_b1[e * 256 + h] * exp_w2[e * 512 + h * 2 + col];
    ws[WS_OB + gid] = s;
  }
}

// ---------------------------------------------------------------------------
// Kernel 1: emotion head (pooled @ emo_w + emo_b) and conc row assembly.
// One block per utterance row.
// ---------------------------------------------------------------------------
__global__ void k_emotion_conc(const float* __restrict__ pooled,
                               const int* __restrict__ spk,
                               const float* __restrict__ emo_w,
                               const float* __restrict__ emo_b,
                               float* __restrict__ ws,
                               float* __restrict__ out) {
  int u   = blockIdx.x;
  int tid = threadIdx.x;
  float* conc = ws + WS_CONC + (size_t)u * FEAT;
  const float* pr = pooled + (size_t)u * HH;

  for (int j = tid; j < HH; j += blockDim.x) conc[j] = pr[j];

  __shared__ float red[128];
  for (int j = 0; j < NEV; ++j) {
    float s = 0.0f;
    for (int kx = tid; kx < HH; kx += blockDim.x) s += pr[kx] * emo_w[kx * NEV + j];
    red[tid] = s;
    __syncthreads();
    for (int off = 64; off > 0; off >>= 1) {
      if (tid < off) red[tid] += red[tid + off];
      __syncthreads();
    }
    if (tid == 0) {
      float v = red[0] + emo_b[j];
      conc[HH + j]     = v;
      out[u * NEV + j] = v;   // emotion_pred output
    }
    __syncthreads();
  }
  if (tid == 0) conc[FEAT - 1] = (float)spk[u];
}

// ---------------------------------------------------------------------------
// Kernel 2: P[1024,16] = conc[1024,776] @ Wc[776,16] via V_WMMA_F32_16X16X4_F32.
// One wave (32 threads) per 16-row tile; K stepped by 4 (194 WMMAs).
// A layout (16x4 f32): lanes 0-15 hold {K=0,K=1}, lanes 16-31 hold {K=2,K=3}.
// B layout (4x16 f32): mirrored (lane = N, half selects K pair).
// C/D layout: VGPR i -> M=i (lanes 0-15) / M=i+8 (lanes 16-31), N = lane&15.
// ---------------------------------------------------------------------------
__global__ void k_wmma_proj(float* __restrict__ ws) {
  const float* X   = ws + WS_CONC;
  const float* WcT = ws + WS_WCT;
  float*       P   = ws + WS_P;

  int lane = threadIdx.x;        // 0..31, one full wave32
  int half = lane >> 4;          // 0 or 1
  int l15  = lane & 15;
  int mb   = blockIdx.x * 16;
  int row  = mb + l15;
  int koff = half * 2;

  const float* xrow = X + (size_t)row * FEAT + koff;
  const float* wcol = WcT + (size_t)l15 * FEAT + koff;

  v8f acc = {};
  for (int kk = 0; kk < FEAT; kk += 4) {
    float2 av = *(const float2*)(xrow + kk);
    float2 bv = *(const float2*)(wcol + kk);
    v2f a; a[0] = av.x; a[1] = av.y;
    v2f b; b[0] = bv.x; b[1] = bv.y;
    acc = __builtin_amdgcn_wmma_f32_16x16x4_f32(
        /*neg_a=*/false, a, /*neg_b=*/false, b,
        /*c_mod=*/(short)0, acc, /*reuse_a=*/false, /*reuse_b=*/false);
  }

  int outrow = mb + half * 8;
#pragma unroll
  for (int i = 0; i < 8; ++i) P[(size_t)(outrow + i) * NC + l15] = acc[i];
}

// ---------------------------------------------------------------------------
// Kernel 3: per-pair soft-gated combine (closed-form, ~20 FLOP per pair).
//   gating = [gt(t)+ge(end)+gb], o[e,c] = ot(t)+oe(end)+obias, cause = g·o
// Pair p (end-major tril order): end = triangular-root(p), t = p - end(end+1)/2.
// ---------------------------------------------------------------------------
__global__ void k_pairs(const float* __restrict__ gate_b,
                        const float* __restrict__ ws,
                        float* __restrict__ out) {
  int gid = blockIdx.x * blockDim.x + threadIdx.x;
  if (gid >= NPAIR) return;

  int b = gid / PPB;
  int p = gid % PPB;

  int end = (int)((sqrtf(8.0f * (float)p + 1.0f) - 1.0f) * 0.5f);
  while ((end + 1) * (end + 2) / 2 <= p) ++end;   // fixup for fp rounding
  while (end * (end + 1) / 2 > p) --end;
  int t = p - end * (end + 1) / 2;

  const float* P  = ws + WS_P;
  const float* ob = ws + WS_OB;
  const float* Pt = P + (size_t)(b * DD + t)   * NC;
  const float* Pe = P + (size_t)(b * DD + end) * NC;

  float g0 = Pt[0] + Pe[2] + gate_b[0];
  float g1 = Pt[1] + Pe[3] + gate_b[1];

  float o00 = Pt[4] + Pe[8]  + ob[0];   // expert 0, class 0
  float o01 = Pt[5] + Pe[9]  + ob[1];   // expert 0, class 1
  float o10 = Pt[6] + Pe[10] + ob[2];   // expert 1, class 0
  float o11 = Pt[7] + Pe[11] + ob[3];   // expert 1, class 1

  float c0 = g0 * o00 + g1 * o10;
  float c1 = g0 * o01 + g1 * o11;

  out[NU * NEV + gid * 2 + 0] = c0;
  out[NU * NEV + gid * 2 + 1] = c1;
}

// ---------------------------------------------------------------------------
extern "C" void kernel_launch(void* const* d_in, const int* in_sizes, int n_in,
                              void* d_out, int out_size, void* d_ws, size_t ws_size,
                              hipStream_t stream) {
  const float* pooled = (const float*)d_in[0];
  const int*   spk    = (const int*)  d_in[1];
  const float* emo_w  = (const float*)d_in[2];
  const float* emo_b  = (const float*)d_in[3];
  const float* gate_w = (const float*)d_in[4];
  const float* gate_b = (const float*)d_in[5];
  const float* exp_w1 = (const float*)d_in[6];
  const float* exp_b1 = (const float*)d_in[7];
  const float* exp_w2 = (const float*)d_in[8];
  const float* exp_b2 = (const float*)d_in[9];
  float* out = (float*)d_out;
  float* ws  = (float*)d_ws;

  (void)in_sizes; (void)n_in; (void)out_size; (void)ws_size;

  // 0) fold expert MLP + gate weights into combined projection columns
  k_build_weights<<<(NC * FEAT + 255) / 256, 256, 0, stream>>>(
      gate_w, exp_w1, exp_b1, exp_w2, exp_b2, ws);

  // 1) emotion head + conc assembly (one block per utterance)
  k_emotion_conc<<<NU, 128, 0, stream>>>(pooled, spk, emo_w, emo_b, ws, out);

  // 2) WMMA projection: conc @ Wc -> P  (64 waves, one 16x16 tile each)
  k_wmma_proj<<<NU / 16, 32, 0, stream>>>(ws);

  // 3) per-pair gated combine -> cause_pred
  k_pairs<<<(NPAIR + 255) / 256, 256, 0, stream>>>(gate_b, ws, out);
}